// MultiTaskDecoder_44418551775376
// MI455X (gfx1250) — compile-verified
//
#include <hip/hip_runtime.h>
#include <stdint.h>

// Problem constants (match reference)
#define B_      2
#define N_      2048
#define C_      8
#define TOP_K   4
#define M_CAND  64          // top candidates handled by the WMMA distance block
#define THREADS 256

static constexpr float CONF_THR  = 0.5f;
static constexpr float DIST_THR2 = 20.0f * 20.0f;

typedef __attribute__((ext_vector_type(2))) float v2f;
typedef __attribute__((ext_vector_type(8))) float v8f;

// One workgroup per (batch, class). 256 threads = 8 wave32.
__global__ __launch_bounds__(THREADS)
void decoder_kernel(const float* __restrict__ loc_g,   // [B, N, 2]
                    const float* __restrict__ cls_g,   // [B, N, C]
                    float* __restrict__ out)           // sel_s[B,C,K] | sel_p[B,C,K,2] | ok[B,C,K]
{
    __shared__ float s_score[N_];
    __shared__ int   s_idx[N_];
    __shared__ float s_pxy[N_ * 2];          // interleaved {x,y}, async-copied image of loc[b]
    __shared__ float s_d2[M_CAND][M_CAND];   // pairwise squared distances of top-64
    __shared__ float s_cx[M_CAND], s_cy[M_CAND], s_cn[M_CAND], s_cs[M_CAND];

    const int wg  = blockIdx.x;          // 0..15
    const int b   = wg / C_;
    const int c   = wg % C_;
    const int tid = threadIdx.x;

    const float* loc = loc_g + (size_t)b * N_ * 2;
    const float* cls = cls_g + (size_t)b * N_ * C_;

    // ---- 0) async-copy this batch's points into LDS (16 KB), bypassing VGPRs ----
    // global_load_async_to_lds_b128 (GVS form): each lane copies 16 B from global saddr+vaddr+off to LDS at vdst+off
    // 256 lanes x 16B = 4 KB per issue; 4 issues with offset: immediates.
    {
        const uint32_t lds_off = (uint32_t)(uintptr_t)(&s_pxy[0]) + (uint32_t)tid * 16u;
        const uint32_t g_off   = (uint32_t)tid * 16u;
        const uint64_t g_base  = (uint64_t)(uintptr_t)loc;
        asm volatile(
            "global_load_async_to_lds_b128 %0, %1, %2\n\t"
            "global_load_async_to_lds_b128 %0, %1, %2 offset:4096\n\t"
            "global_load_async_to_lds_b128 %0, %1, %2 offset:8192\n\t"
            "global_load_async_to_lds_b128 %0, %1, %2 offset:12288\n\t"
            "s_wait_asynccnt 0x0"
            :
            : "v"(lds_off), "v"(g_off), "s"(g_base)
            : "memory");
    }

    // ---- 1) softmax score for class c ----
    for (int n = tid; n < N_; n += THREADS) {
        float l[C_];
        float mx = -3.0e38f;
        #pragma unroll
        for (int k = 0; k < C_; ++k) { l[k] = cls[n * C_ + k]; mx = fmaxf(mx, l[k]); }
        float sum = 0.0f;
        #pragma unroll
        for (int k = 0; k < C_; ++k) sum += __expf(l[k] - mx);
        s_score[n] = __expf(l[c] - mx) / sum;
        s_idx[n]   = n;
    }
    __syncthreads();

    // ---- 2) bitonic sort: score descending, index ascending on ties ----
    // (matches stable jnp.argsort(-scores))
    for (int k = 2; k <= N_; k <<= 1) {
        for (int j = k >> 1; j > 0; j >>= 1) {
            for (int i = tid; i < N_; i += THREADS) {
                const int ixj = i ^ j;
                if (ixj > i) {
                    const bool dir = ((i & k) == 0);       // forward = desired order
                    const float sa = s_score[i], sb = s_score[ixj];
                    const int   ia = s_idx[i],   ib = s_idx[ixj];
                    const bool before = (sa > sb) || (sa == sb && ia < ib);
                    if (before != dir) {
                        s_score[i] = sb; s_score[ixj] = sa;
                        s_idx[i]   = ib; s_idx[ixj]   = ia;
                    }
                }
            }
            __syncthreads();
        }
    }

    // ---- 3) gather top-64 candidates (already in sorted order) ----
    if (tid < M_CAND) {
        const int oi = s_idx[tid];
        const float x = s_pxy[2 * oi + 0], y = s_pxy[2 * oi + 1];
        s_cs[tid] = s_score[tid];
        s_cx[tid] = x;
        s_cy[tid] = y;
        s_cn[tid] = x * x + y * y;        // ||p||^2 for Gram expansion
    }
    __syncthreads();

    // ---- 4) WMMA: 64x64 squared-distance block via Gram matrix ----
    // D2[m][n] = n2[m] + n2[n] - 2 * (x_m x_n + y_m y_n)
    // Cross term = P (64x2) * P^T (2x64), tiled as V_WMMA_F32_16X16X4_F32 with
    // K=2 real data; K=2,3 (lanes 16-31 per ISA 32-bit A/B layout) are zero.
    {
        const int wave = tid >> 5;        // 0..7
        const int lane = tid & 31;
        for (int t = wave; t < 16; t += 8) {      // 4x4 tile grid, 2 tiles/wave
            const int tm = t >> 2, tn = t & 3;
            v2f afrag = {0.0f, 0.0f};
            v2f bfrag = {0.0f, 0.0f};
            if (lane < 16) {
                // A: lane L<16 holds row M=tm*16+L, VGPR0=K0(x), VGPR1=K1(y)
                afrag.x = s_cx[tm * 16 + lane];
                afrag.y = s_cy[tm * 16 + lane];
                // B: lane L<16 holds col N=tn*16+L, VGPR0=K0(x), VGPR1=K1(y)
                bfrag.x = s_cx[tn * 16 + lane];
                bfrag.y = s_cy[tn * 16 + lane];
            }
            v8f acc = {};
            acc = __builtin_amdgcn_wmma_f32_16x16x4_f32(
                      /*neg_a=*/false, afrag,
                      /*neg_b=*/false, bfrag,
                      /*c_mod=*/(short)0, acc,
                      /*reuse_a=*/false, /*reuse_b=*/false);
            // C/D layout: VGPR r -> (M = r [+8 for hi lanes], N = lane % 16)
            const int nn    = (lane < 16) ? lane : (lane - 16);
            const int mbase = (lane < 16) ? 0 : 8;
            #pragma unroll
            for (int r = 0; r < 8; ++r) {
                const int m = tm * 16 + mbase + r;
                const int n = tn * 16 + nn;
                s_d2[m][n] = s_cn[m] + s_cn[n] - 2.0f * acc[r];
            }
        }
    }
    __syncthreads();

    // ---- 5) greedy NMS (only kept points suppress; stop at TOP_K) ----
    if (tid == 0) {
        float kx[TOP_K], ky[TOP_K], ks[TOP_K];
        int   krow[TOP_K];
        int   kcnt  = 0;
        bool  below = false;

        // Phase 1: top-64 using the WMMA distance block
        for (int r = 0; r < M_CAND && kcnt < TOP_K; ++r) {
            const float sc = s_cs[r];
            if (!(sc > CONF_THR)) { below = true; break; }   // sorted desc -> done
            bool keep = true;
            for (int k = 0; k < kcnt; ++k)
                if (s_d2[krow[k]][r] <= DIST_THR2) { keep = false; break; }
            if (keep) {
                krow[kcnt] = r;
                kx[kcnt] = s_cx[r]; ky[kcnt] = s_cy[r]; ks[kcnt] = sc;
                ++kcnt;
            }
        }
        // Phase 2 (rare): continue past rank 64 if needed
        if (!below && kcnt < TOP_K) {
            for (int r = M_CAND; r < N_ && kcnt < TOP_K; ++r) {
                const float sc = s_score[r];
                if (!(sc > CONF_THR)) break;
                const int oi = s_idx[r];
                const float x = s_pxy[2 * oi + 0], y = s_pxy[2 * oi + 1];
                bool keep = true;
                for (int k = 0; k < kcnt; ++k) {
                    const float dx = x - kx[k], dy = y - ky[k];
                    if (dx * dx + dy * dy <= DIST_THR2) { keep = false; break; }
                }
                if (keep) { kx[kcnt] = x; ky[kcnt] = y; ks[kcnt] = sc; ++kcnt; }
            }
        }

        // ---- 6) outputs, concatenated flat: sel_s | sel_p | ok ----
        float* out_s  = out;                          // [B,C,K]
        float* out_p  = out + B_ * C_ * TOP_K;        // [B,C,K,2]
        float* out_ok = out + B_ * C_ * TOP_K * 3;    // [B,C,K]
        const int base = (b * C_ + c) * TOP_K;
        #pragma unroll
        for (int k = 0; k < TOP_K; ++k) {
            const bool v = (k < kcnt);
            out_s[base + k]           = v ? ks[k] : 0.0f;
            out_p[(base + k) * 2 + 0] = v ? kx[k] : 0.0f;
            out_p[(base + k) * 2 + 1] = v ? ky[k] : 0.0f;
            out_ok[base + k]          = v ? 1.0f : 0.0f;
        }
    }
}

extern "C" void kernel_launch(void* const* d_in, const int* in_sizes, int n_in,
                              void* d_out, int out_size, void* d_ws, size_t ws_size,
                              hipStream_t stream) {
    (void)in_sizes; (void)n_in; (void)out_size; (void)d_ws; (void)ws_size;
    const float* locations       = (const float*)d_in[0];   // [B,N,2] f32
    const float* classifications = (const float*)d_in[1];   // [B,N,C] f32
    float* out = (float*)d_out;                             // 256 floats total

    decoder_kernel<<<dim3(B_ * C_), dim3(THREADS), 0, stream>>>(
        locations, classifications, out);
}